// GaussianKernelDensity_17282948399819
// MI455X (gfx1250) — compile-verified
//
#include <hip/hip_runtime.h>

typedef __attribute__((ext_vector_type(2))) float v2f;
typedef __attribute__((ext_vector_type(4))) float v4f;
typedef __attribute__((ext_vector_type(8))) float v8f;

#define M_ROWS 4096
#define N_COLS 50000
#define DIM    64
#define WAVES  8
#define TILE_N 16

// coeff = -log(N) - log(2*pi)*D/2 - log(1.0)*D  (frozen at init per reference)
#define KDE_COEFF (-69.63184405430771f)

__global__ __launch_bounds__(256)
void kde_logsumexp_wmma(const float* __restrict__ x,
                        const float* __restrict__ Xm,
                        const float* __restrict__ bw,
                        float* __restrict__ out) {
  __shared__ float lds_x2[16];
  __shared__ float lds_m[WAVES][16];
  __shared__ float lds_s[WAVES][16];

  const int tid  = threadIdx.x;
  const int wave = tid >> 5;          // wave32
  const int lane = tid & 31;
  const int hi   = lane >> 4;         // half-wave: 0 or 1
  const int l16  = lane & 15;
  const int row0 = blockIdx.x * 16;   // this block's M tile

  // ||x_row||^2 for the 16 rows of this tile (one-time; x tile is 4KB, L2-hot)
  if (tid < 16) {
    const float* xr = x + (size_t)(row0 + tid) * DIM;
    float acc = 0.0f;
#pragma unroll
    for (int d = 0; d < DIM; ++d) acc = fmaf(xr[d], xr[d], acc);
    lds_x2[tid] = acc;
  }
  __syncthreads();

  // A fragments, register-resident across the entire N loop.
  // K-permutation: chunk c = float4 at element 8c + 4*hi; .xy feeds WMMA step 2c,
  // .zw feeds step 2c+1. A and B use the identical permutation, so the K-sum is exact.
  v4f a4[8];
  {
    const float* xr = x + (size_t)(row0 + l16) * DIM + 4 * hi;
#pragma unroll
    for (int c = 0; c < 8; ++c) a4[c] = *(const v4f*)(xr + 8 * c);
  }

  const float b  = bw[0];
  const float s2 = -1.4426950408889634f / (2.0f * b * b); // log2-domain scale (<0)
  const float u  = -2.0f * s2;                            // > 0

  // per-slot row constants: x2[row]*s2 (C vgpr k holds row k + 8*hi)
  float x2s2[8];
#pragma unroll
  for (int k = 0; k < 8; ++k) x2s2[k] = lds_x2[8 * hi + k] * s2;

  // fixed-shift accumulation: logits <= 0 -> sum of exp2(l) in [0, N]; no rescale.
  // m[k] tracks the max logit purely as an underflow fallback (1 VALU/elem).
  float m[8], s[8];
#pragma unroll
  for (int k = 0; k < 8; ++k) { m[k] = -1.0e30f; s[k] = 0.0f; }

  // base pointer for this lane's B-fragment column slice
  const float* Xbase = Xm + (size_t)l16 * DIM + 4 * hi;

  auto loadB = [&](v4f (&bf)[8], int t) {
    const float* Xc = Xbase + (size_t)t * (TILE_N * DIM);
#pragma unroll
    for (int c = 0; c < 8; ++c) bf[c] = *(const v4f*)(Xc + 8 * c);
  };

  auto computeTile = [&](const v4f (&bf)[8]) {
    // ||X_col||^2 from the fragments (pk-FMA friendly); halves combine via shuffle
    float p0 = 0.f, p1 = 0.f, p2 = 0.f, p3 = 0.f;
#pragma unroll
    for (int c = 0; c < 8; ++c) {
      p0 = fmaf(bf[c].x, bf[c].x, p0);
      p1 = fmaf(bf[c].y, bf[c].y, p1);
      p2 = fmaf(bf[c].z, bf[c].z, p2);
      p3 = fmaf(bf[c].w, bf[c].w, p3);
    }
    const float p     = (p0 + p1) + (p2 + p3);
    const float X2col = p + __shfl_xor(p, 16, 32);

    // 16 x V_WMMA_F32_16X16X4_F32 over D=64, split into two independent
    // accumulation chains so the XDL pipe can overlap WMMA latency.
    v8f c0 = {0.f, 0.f, 0.f, 0.f, 0.f, 0.f, 0.f, 0.f};
    v8f c1 = {0.f, 0.f, 0.f, 0.f, 0.f, 0.f, 0.f, 0.f};
#pragma unroll
    for (int c = 0; c < 4; ++c) {
      const v2f alo0 = __builtin_shufflevector(a4[c],     a4[c],     0, 1);
      const v2f ahi0 = __builtin_shufflevector(a4[c],     a4[c],     2, 3);
      const v2f blo0 = __builtin_shufflevector(bf[c],     bf[c],     0, 1);
      const v2f bhi0 = __builtin_shufflevector(bf[c],     bf[c],     2, 3);
      const v2f alo1 = __builtin_shufflevector(a4[c + 4], a4[c + 4], 0, 1);
      const v2f ahi1 = __builtin_shufflevector(a4[c + 4], a4[c + 4], 2, 3);
      const v2f blo1 = __builtin_shufflevector(bf[c + 4], bf[c + 4], 0, 1);
      const v2f bhi1 = __builtin_shufflevector(bf[c + 4], bf[c + 4], 2, 3);
      c0 = __builtin_amdgcn_wmma_f32_16x16x4_f32(false, alo0, false, blo0,
                                                 (short)0, c0, false, false);
      c1 = __builtin_amdgcn_wmma_f32_16x16x4_f32(false, alo1, false, blo1,
                                                 (short)0, c1, false, false);
      c0 = __builtin_amdgcn_wmma_f32_16x16x4_f32(false, ahi0, false, bhi0,
                                                 (short)0, c0, false, false);
      c1 = __builtin_amdgcn_wmma_f32_16x16x4_f32(false, ahi1, false, bhi1,
                                                 (short)0, c1, false, false);
    }
    const v8f cc = c0 + c1;   // 4 x v_pk_add_f32

    // epilogue: l = (x2 + X2 - 2*dot)*s2 = fma(dot, u, rc); clamp <= 0; 1 exp2/elem
    const float base = X2col * s2;
#pragma unroll
    for (int k = 0; k < 8; ++k) {
      const float rc = x2s2[k] + base;
      float l = fmaf(cc[k], u, rc);
      l = fminf(l, 0.0f);                 // == max(d2,0)*s2 since s2 < 0
      m[k] = fmaxf(m[k], l);
      s[k] += __builtin_amdgcn_exp2f(l);
    }
  };

  // --- double-buffered main loop over N tiles (ping-pong, no register copies) ---
  const int ntiles = N_COLS / TILE_N;   // 3125, exact; every wave gets >= 390 tiles
  {
    v4f b0[8], b1[8];
    int t = wave;
    loadB(b0, t);
    for (;;) {
      const int t1 = t + WAVES;
      if (t1 >= ntiles) { computeTile(b0); break; }
      loadB(b1, t1);               // tile t+8 loads fly over tile t's compute
      computeTile(b0);
      const int t2 = t1 + WAVES;
      if (t2 >= ntiles) { computeTile(b1); break; }
      loadB(b0, t2);               // tile t+16 loads fly over tile t+8's compute
      computeTile(b1);
      t = t2;
    }
  }

  // reduce across the 16 lanes of each half-wave (same rows, different N columns)
#pragma unroll
  for (int off = 1; off < 16; off <<= 1) {
#pragma unroll
    for (int k = 0; k < 8; ++k) {
      s[k] += __shfl_xor(s[k], off, 32);
      m[k] = fmaxf(m[k], __shfl_xor(m[k], off, 32));
    }
  }

  if (l16 == 0) {
#pragma unroll
    for (int k = 0; k < 8; ++k) {
      lds_m[wave][8 * hi + k] = m[k];
      lds_s[wave][8 * hi + k] = s[k];
    }
  }
  __syncthreads();

  // merge the 8 waves' partials and emit
  if (tid < 16) {
    float ss = 0.0f, mm = -1.0e30f;
#pragma unroll
    for (int w = 0; w < WAVES; ++w) {
      ss += lds_s[w][tid];
      mm = fmaxf(mm, lds_m[w][tid]);
    }
    // log2-domain result; if the sum fully underflowed (astronomically rare),
    // fall back to the max logit (exact limit of logsumexp under underflow).
    const float r = (ss > 0.0f) ? __builtin_amdgcn_logf(ss) : mm;
    out[row0 + tid] = r * 0.6931471805599453f + KDE_COEFF;
  }
}

extern "C" void kernel_launch(void* const* d_in, const int* in_sizes, int n_in,
                              void* d_out, int out_size, void* d_ws, size_t ws_size,
                              hipStream_t stream) {
  const float* x  = (const float*)d_in[0];
  const float* Xm = (const float*)d_in[1];
  const float* bw = (const float*)d_in[2];
  float* outp = (float*)d_out;

  dim3 grid(M_ROWS / 16);   // 256 workgroups, one 16-row tile each
  dim3 block(256);          // 8 wave32 waves
  hipLaunchKernelGGL(kde_logsumexp_wmma, grid, block, 0, stream, x, Xm, bw, outp);
}